// CGCNN_Model_46772193853678
// MI455X (gfx1250) — compile-verified
//
#include <hip/hip_runtime.h>
#include <math.h>

#define NNODES 30000
#define NEDGES 480000
#define NGRAPHS 300
#define ND 64
#define ED 41
#define FCD 128
#define NCONV 6
#define NFCHID 3
#define EPSV 1e-5f

#define KDIM (2*ND + ED)   /* 169 */
#define KPAD 172
#define KSTEPS 43          /* KPAD/4 */

typedef float v2f __attribute__((ext_vector_type(2)));
typedef float v8f __attribute__((ext_vector_type(8)));

__device__ __forceinline__ float softplusf(float x) {
    return fmaxf(x, 0.0f) + log1pf(expf(-fabsf(x)));
}
__device__ __forceinline__ float sigmoidf(float x) {
    return 1.0f / (1.0f + expf(-x));
}

// ---------------- Embedding: nf = x @ embW + embB  (30000x92 @ 92x64) ---------
__global__ void k_embed(const float* __restrict__ x, const float* __restrict__ W,
                        const float* __restrict__ b, float* __restrict__ nf) {
    int t = blockIdx.x * 256 + threadIdx.x;     // 30000*64 threads
    int n = t >> 6, f = t & 63;
    const float* xr = x + (size_t)n * 92;
    float s = b[f];
    #pragma unroll 4
    for (int k = 0; k < 92; ++k) s = fmaf(xr[k], W[k * ND + f], s);
    nf[t] = s;
}

// ---------------- CSR build over dst ------------------------------------------
__global__ void k_count(const int* __restrict__ ei, int* __restrict__ counts) {
    int e = blockIdx.x * 256 + threadIdx.x;
    atomicAdd(&counts[ei[NEDGES + e]], 1);
}

__global__ void k_scan(const int* __restrict__ counts, int* __restrict__ rowptr,
                       int* __restrict__ cursor) {
    __shared__ int lds[1024];
    int tid = threadIdx.x;
    int start = tid * 30;                       // 1024*30 = 30720 >= 30000
    int local = 0;
    for (int i = 0; i < 30; ++i) {
        int idx = start + i;
        if (idx < NNODES) local += counts[idx];
    }
    lds[tid] = local;
    __syncthreads();
    for (int off = 1; off < 1024; off <<= 1) {
        int cur = lds[tid];
        int add = (tid >= off) ? lds[tid - off] : 0;
        __syncthreads();
        lds[tid] = cur + add;
        __syncthreads();
    }
    int run = lds[tid] - local;                 // exclusive prefix
    for (int i = 0; i < 30; ++i) {
        int idx = start + i;
        if (idx < NNODES) {
            rowptr[idx] = run;
            cursor[idx] = run;
            run += counts[idx];
        }
    }
    if (tid == 1023) rowptr[NNODES] = lds[1023];
}

__global__ void k_fill(const int* __restrict__ ei, int* __restrict__ cursor,
                       int* __restrict__ eids) {
    int e = blockIdx.x * 256 + threadIdx.x;
    int pos = atomicAdd(&cursor[ei[NEDGES + e]], 1);
    eids[pos] = e;
}

// ---------------- Edge GEMM: z = [nf[dst]|nf[src]|ea] @ W, + BN stats ---------
// block = 256 threads (8 waves); each wave does a 16(edge) x 128 tile with
// 8 x V_WMMA_F32_16X16X4_F32 accumulators; K = 169 (padded 172), 43 k-steps.
__global__ __launch_bounds__(256)
void k_edge_gemm(const float* __restrict__ nf, const float* __restrict__ ea,
                 const int* __restrict__ ei, const float* __restrict__ Wg,
                 float* __restrict__ z, float* __restrict__ gsum,
                 float* __restrict__ gq) {
    extern __shared__ char smem[];
    v2f*   Wf   = (v2f*)smem;                               // [43][8][32] B-frags
    float* G    = (float*)(smem + KSTEPS * 8 * 32 * 8);     // [128][KPAD]
    float* lsum = (float*)(smem + KSTEPS * 8 * 32 * 8 + 128 * KPAD * 4);
    float* lq   = lsum + 128;
    int*   sdst = (int*)(lq + 128);
    int*   ssrc = sdst + 128;

    const int tid = threadIdx.x;
    const int eb  = blockIdx.x * 128;

    if (tid < 128) {
        sdst[tid] = ei[NEDGES + eb + tid];
        ssrc[tid] = ei[eb + tid];
        lsum[tid] = 0.0f;
        lq[tid]   = 0.0f;
    }
    __syncthreads();

    // Swizzle W (row-major [K][128]) into per-lane B fragment layout:
    // frag(ks,j): lane L -> n = j*16+(L&15), kk = ks*4+(L>>4)*2; {W[kk][n],W[kk+1][n]}
    for (int idx = tid; idx < KSTEPS * 8 * 32; idx += 256) {
        int lane = idx & 31;
        int j    = (idx >> 5) & 7;
        int ks   = idx >> 8;
        int kk   = ks * 4 + ((lane >> 4) << 1);
        int n    = j * 16 + (lane & 15);
        v2f w;
        w.x = (kk     < KDIM) ? Wg[(size_t)kk * FCD + n]       : 0.0f;
        w.y = (kk + 1 < KDIM) ? Wg[(size_t)(kk + 1) * FCD + n] : 0.0f;
        Wf[idx] = w;
    }
    // Stage gathered A tile: 128 edges x KPAD (zero pad K 169..171)
    for (int idx = tid; idx < 128 * KPAD; idx += 256) {
        int el = idx / KPAD;
        int c  = idx - el * KPAD;
        float v;
        if      (c < ND)     v = nf[(size_t)sdst[el] * ND + c];
        else if (c < 2 * ND) v = nf[(size_t)ssrc[el] * ND + (c - ND)];
        else if (c < KDIM)   v = ea[(size_t)(eb + el) * ED + (c - 2 * ND)];
        else                 v = 0.0f;
        G[idx] = v;
    }
    __syncthreads();

    const int lane = tid & 31;
    const int wv   = tid >> 5;
    const int mrow = wv * 16 + (lane & 15);
    const int ksel = (lane >> 4) << 1;

    v8f acc[8];
    #pragma unroll
    for (int j = 0; j < 8; ++j)
        #pragma unroll
        for (int r = 0; r < 8; ++r) acc[j][r] = 0.0f;

    const float* Grow = G + mrow * KPAD;
    #pragma unroll 1
    for (int ks = 0; ks < KSTEPS; ++ks) {
        v2f a = *(const v2f*)(Grow + ks * 4 + ksel);         // 8B-aligned ds_load_b64
        const v2f* wrow = Wf + ks * 8 * 32 + lane;
        #pragma unroll
        for (int j = 0; j < 8; ++j) {
            acc[j] = __builtin_amdgcn_wmma_f32_16x16x4_f32(
                false, a, false, wrow[j * 32], (short)0, acc[j], false, false);
        }
    }

    // Store z and accumulate BN statistics (block-local LDS, then global atomics)
    #pragma unroll
    for (int j = 0; j < 8; ++j) {
        int n = j * 16 + (lane & 15);
        float s = 0.0f, q = 0.0f;
        #pragma unroll
        for (int r = 0; r < 8; ++r) {
            float v = acc[j][r];
            s += v;
            q += v * v;
            int m = ((lane >> 4) << 3) + r;                  // D row within 16
            z[(size_t)(eb + wv * 16 + m) * FCD + n] = v;
        }
        atomicAdd(&lsum[n], s);
        atomicAdd(&lq[n], q);
    }
    __syncthreads();
    if (tid < 128) {
        atomicAdd(&gsum[tid], lsum[tid]);
        atomicAdd(&gq[tid],   lq[tid]);
    }
}

// ---------------- BN stat finalize --------------------------------------------
__global__ void k_bn_finalize(const float* __restrict__ gsum, const float* __restrict__ gq,
                              float* __restrict__ mu, float* __restrict__ rstd) {
    int i = threadIdx.x;
    float m = gsum[i] * (1.0f / (float)NEDGES);
    float v = gq[i] * (1.0f / (float)NEDGES) - m * m;
    v = fmaxf(v, 0.0f);
    mu[i]   = m;
    rstd[i] = rsqrtf(v + EPSV);
}

// ------------- Fused gate + segment-sum + LayerNorm + residual softplus -------
// One wave per node; lane handles features {lane, lane+32}.
__global__ __launch_bounds__(256)
void k_agg_ln(const float* __restrict__ z, const int* __restrict__ rowptr,
              const int* __restrict__ eids, const float* __restrict__ mu,
              const float* __restrict__ rstd, const float* __restrict__ bg,
              const float* __restrict__ bb, const float* __restrict__ lg,
              const float* __restrict__ lb, const float* __restrict__ nfin,
              float* __restrict__ nfout) {
    int lane = threadIdx.x & 31;
    int node = blockIdx.x * 8 + (threadIdx.x >> 5);
    int c0 = lane, c1 = lane + 32, c2 = lane + 64, c3 = lane + 96;
    float mu0 = mu[c0], mu1 = mu[c1], mu2c = mu[c2], mu3 = mu[c3];
    float r0 = rstd[c0], r1 = rstd[c1], r2 = rstd[c2], r3 = rstd[c3];
    float g0 = bg[c0], g1 = bg[c1], g2 = bg[c2], g3 = bg[c3];
    float b0 = bb[c0], b1 = bb[c1], b2 = bb[c2], b3 = bb[c3];

    float a0 = 0.0f, a1 = 0.0f;
    int pend = rowptr[node + 1];
    for (int p = rowptr[node]; p < pend; ++p) {
        const float* zr = z + (size_t)eids[p] * FCD;
        float v0 = fmaf((zr[c0] - mu0)  * r0, g0, b0);
        float v1 = fmaf((zr[c1] - mu1)  * r1, g1, b1);
        float v2 = fmaf((zr[c2] - mu2c) * r2, g2, b2);
        float v3 = fmaf((zr[c3] - mu3)  * r3, g3, b3);
        a0 += sigmoidf(v0) * softplusf(v2);
        a1 += sigmoidf(v1) * softplusf(v3);
    }
    // LayerNorm over the 64 aggregated features (wave reduction, 2 per lane)
    float s = a0 + a1, q = a0 * a0 + a1 * a1;
    for (int off = 16; off > 0; off >>= 1) {
        s += __shfl_xor(s, off, 32);
        q += __shfl_xor(q, off, 32);
    }
    float m2 = s * (1.0f / 64.0f);
    float var2 = fmaxf(q * (1.0f / 64.0f) - m2 * m2, 0.0f);
    float rs = rsqrtf(var2 + EPSV);
    float ln0 = fmaf((a0 - m2) * rs, lg[lane],      lb[lane]);
    float ln1 = fmaf((a1 - m2) * rs, lg[lane + 32], lb[lane + 32]);
    size_t base = (size_t)node * ND;
    nfout[base + lane]      = softplusf(ln0 + nfin[base + lane]);
    nfout[base + lane + 32] = softplusf(ln1 + nfin[base + lane + 32]);
}

// ---------------- Graph pooling + FC head -------------------------------------
__global__ void k_pool(const float* __restrict__ nf, const int* __restrict__ batch,
                       float* __restrict__ pool, float* __restrict__ cnt) {
    int t = blockIdx.x * 256 + threadIdx.x;     // 30000*64
    int n = t >> 6, f = t & 63;
    int g = batch[n];
    atomicAdd(&pool[g * ND + f], nf[t]);
    if (f == 0) atomicAdd(&cnt[g], 1.0f);
}

__global__ void k_fc1(const float* __restrict__ pool, const float* __restrict__ cnt,
                      const float* __restrict__ W, const float* __restrict__ b,
                      float* __restrict__ h) {
    int t = blockIdx.x * 256 + threadIdx.x;     // 300*128
    int g = t >> 7, j = t & 127;
    float inv = 1.0f / fmaxf(cnt[g], 1.0f);
    const float* pr = pool + g * ND;
    float s = b[j];
    #pragma unroll 4
    for (int k = 0; k < ND; ++k) s = fmaf(pr[k] * inv, W[k * FCD + j], s);
    h[t] = softplusf(s);
}

__global__ void k_fch(const float* __restrict__ hin, const float* __restrict__ W,
                      const float* __restrict__ b, float* __restrict__ hout) {
    int t = blockIdx.x * 256 + threadIdx.x;     // 300*128
    int g = t >> 7, j = t & 127;
    const float* hr = hin + g * FCD;
    float s = b[j];
    #pragma unroll 4
    for (int k = 0; k < FCD; ++k) s = fmaf(hr[k], W[k * FCD + j], s);
    hout[t] = softplusf(s);
}

__global__ void k_fco(const float* __restrict__ h, const float* __restrict__ W,
                      const float* __restrict__ b, float* __restrict__ out) {
    int g = blockIdx.x * 256 + threadIdx.x;
    if (g >= NGRAPHS) return;
    const float* hr = h + g * FCD;
    float s = b[0];
    for (int k = 0; k < FCD; ++k) s = fmaf(hr[k], W[k], s);
    out[g] = s;
}

// ---------------- Host launcher -----------------------------------------------
extern "C" void kernel_launch(void* const* d_in, const int* in_sizes, int n_in,
                              void* d_out, int out_size, void* d_ws, size_t ws_size,
                              hipStream_t stream) {
    const float* x     = (const float*)d_in[0];
    const float* ea    = (const float*)d_in[1];
    const float* embW  = (const float*)d_in[2];
    const float* embB  = (const float*)d_in[3];
    const float* convW = (const float*)d_in[4];
    // d_in[5] = convB: cancels inside BatchNorm (mean subtraction) — unused.
    const float* bnG   = (const float*)d_in[6];
    const float* bnB   = (const float*)d_in[7];
    const float* lnG   = (const float*)d_in[8];
    const float* lnB   = (const float*)d_in[9];
    const float* fc1W  = (const float*)d_in[10];
    const float* fc1B  = (const float*)d_in[11];
    const float* fcsW  = (const float*)d_in[12];
    const float* fcsB  = (const float*)d_in[13];
    const float* foW   = (const float*)d_in[14];
    const float* foB   = (const float*)d_in[15];
    const int*   ei    = (const int*)d_in[16];
    const int*   batch = (const int*)d_in[17];
    float* out = (float*)d_out;

    char* ws = (char*)d_ws;
    size_t off = 0;
    auto alloc = [&](size_t bytes) -> void* {
        void* p = ws + off;
        off = (off + bytes + 255) & ~(size_t)255;
        return p;
    };
    float* nf_a   = (float*)alloc((size_t)NNODES * ND * 4);
    float* nf_b   = (float*)alloc((size_t)NNODES * ND * 4);
    float* zbuf   = (float*)alloc((size_t)NEDGES * FCD * 4);
    int*   counts = (int*)  alloc((size_t)NNODES * 4);
    int*   cursor = (int*)  alloc((size_t)NNODES * 4);
    int*   rowptr = (int*)  alloc((size_t)(NNODES + 1) * 4);
    int*   eids   = (int*)  alloc((size_t)NEDGES * 4);
    float* gstats = (float*)alloc(2 * FCD * 4);       // gsum | gq (contiguous)
    float* gsum   = gstats;
    float* gq     = gstats + FCD;
    float* mu     = (float*)alloc(FCD * 4);
    float* rstd   = (float*)alloc(FCD * 4);
    float* pool   = (float*)alloc((size_t)(NGRAPHS * ND + NGRAPHS) * 4); // pool|cnt
    float* cnt    = pool + NGRAPHS * ND;
    float* h0     = (float*)alloc((size_t)NGRAPHS * FCD * 4);
    float* h1     = (float*)alloc((size_t)NGRAPHS * FCD * 4);
    (void)in_sizes; (void)n_in; (void)out_size; (void)ws_size;

    const size_t smem_gemm = (size_t)KSTEPS * 8 * 32 * 8    // B fragments
                           + (size_t)128 * KPAD * 4         // A tile
                           + 2 * 128 * 4                    // BN partial sums
                           + 2 * 128 * 4;                   // dst/src ids
    // 1) embedding
    k_embed<<<(NNODES * ND) / 256, 256, 0, stream>>>(x, embW, embB, nf_a);
    // 2) CSR over dst (reused by all 6 layers)
    hipMemsetAsync(counts, 0, (size_t)NNODES * 4, stream);
    k_count<<<NEDGES / 256, 256, 0, stream>>>(ei, counts);
    k_scan<<<1, 1024, 0, stream>>>(counts, rowptr, cursor);
    k_fill<<<NEDGES / 256, 256, 0, stream>>>(ei, cursor, eids);
    // 3) conv layers
    float* cur = nf_a;
    float* nxt = nf_b;
    for (int l = 0; l < NCONV; ++l) {
        hipMemsetAsync(gstats, 0, 2 * FCD * 4, stream);
        k_edge_gemm<<<NEDGES / 128, 256, smem_gemm, stream>>>(
            cur, ea, ei, convW + (size_t)l * KDIM * FCD, zbuf, gsum, gq);
        k_bn_finalize<<<1, FCD, 0, stream>>>(gsum, gq, mu, rstd);
        k_agg_ln<<<NNODES / 8, 256, 0, stream>>>(
            zbuf, rowptr, eids, mu, rstd,
            bnG + (size_t)l * FCD, bnB + (size_t)l * FCD,
            lnG + (size_t)l * ND,  lnB + (size_t)l * ND, cur, nxt);
        float* t = cur; cur = nxt; nxt = t;
    }
    // 4) pooling + FC head
    hipMemsetAsync(pool, 0, (size_t)(NGRAPHS * ND + NGRAPHS) * 4, stream);
    k_pool<<<(NNODES * ND) / 256, 256, 0, stream>>>(cur, batch, pool, cnt);
    k_fc1<<<(NGRAPHS * FCD) / 256, 256, 0, stream>>>(pool, cnt, fc1W, fc1B, h0);
    float* hc = h0;
    float* hn = h1;
    for (int l = 0; l < NFCHID; ++l) {
        k_fch<<<(NGRAPHS * FCD) / 256, 256, 0, stream>>>(
            hc, fcsW + (size_t)l * FCD * FCD, fcsB + (size_t)l * FCD, hn);
        float* t = hc; hc = hn; hn = t;
    }
    k_fco<<<(NGRAPHS + 255) / 256, 256, 0, stream>>>(hc, foW, foB, out);
}